// GAT_39539468927410
// MI455X (gfx1250) — compile-verified
//
#include <hip/hip_runtime.h>

#define NEG_SLOPE 0.2f
#define EPSI 1e-16f

typedef float v2f __attribute__((ext_vector_type(2)));
typedef float v8f __attribute__((ext_vector_type(8)));

// ---------------------------------------------------------------------------
// GEMM: C[N,64] = A[N,K] @ B[K,64] using V_WMMA_F32_16X16X4_F32 (exact fp32).
// 256 threads = 8 waves; each wave owns one 16x16 output tile (2 row-tiles x
// 4 col-tiles per block => 32 rows per block). K stepped by 4.
// Fragment layouts per CDNA5 ISA 7.12.2:
//   A 16x4 : lanes 0-15 -> K=0(v0),1(v1); lanes 16-31 -> K=2(v0),3(v1); M=lane%16
//   B 4x16 : v0: K=0 (lo lanes)/K=2 (hi); v1: K=1/K=3; N=lane%16
//   C/D    : vgpr r: rows r (lo lanes) / r+8 (hi lanes), col = lane%16
// ---------------------------------------------------------------------------
__global__ __launch_bounds__(256) void gat_gemm_wmma(const float* __restrict__ A,
                                                     const float* __restrict__ B,
                                                     float* __restrict__ C,
                                                     int N, int K) {
  const int wave  = threadIdx.x >> 5;
  const int lane  = threadIdx.x & 31;
  const int lhalf = lane & 15;
  const bool hi   = lane >= 16;
  const int m0 = blockIdx.x * 32 + (wave >> 2) * 16;
  const int n0 = (wave & 3) * 16;

  int ar = m0 + lhalf;
  if (ar >= N) ar = N - 1;                 // clamp (N is a multiple of 32 anyway)
  const float* Arow = A + (long long)ar * K + (hi ? 2 : 0);
  const float* Bcol = B + n0 + lhalf + (hi ? 2 * 64 : 0);

  v8f acc = {};
  for (int k0 = 0; k0 < K; k0 += 4) {
    v2f a, b;
    a.x = Arow[k0];
    a.y = Arow[k0 + 1];
    b.x = Bcol[(long long)k0 * 64];
    b.y = Bcol[(long long)(k0 + 1) * 64];
    acc = __builtin_amdgcn_wmma_f32_16x16x4_f32(false, a, false, b,
                                                (short)0, acc, false, false);
  }

  const int row_base = m0 + (hi ? 8 : 0);
  const int col = n0 + lhalf;
#pragma unroll
  for (int r = 0; r < 8; ++r) {
    int row = row_base + r;
    if (row < N) C[(long long)row * 64 + col] = acc[r];
  }
}

// a_s[n,h] = sum_c feat[n,h*8+c]*att_src[h*8+c]; same for a_d. Thread per (n,h).
__global__ __launch_bounds__(256) void gat_attn_coeff(const float* __restrict__ feat,
                                                      const float* __restrict__ att_src,
                                                      const float* __restrict__ att_dst,
                                                      float* __restrict__ a_s,
                                                      float* __restrict__ a_d, int N) {
  int idx = blockIdx.x * blockDim.x + threadIdx.x;  // n*8+h
  if (idx >= N * 8) return;
  int h = idx & 7;
  const float* f = feat + (long long)(idx >> 3) * 64 + h * 8;
  float s = 0.f, d = 0.f;
#pragma unroll
  for (int c = 0; c < 8; ++c) {
    s += f[c] * att_src[h * 8 + c];
    d += f[c] * att_dst[h * 8 + c];
  }
  a_s[idx] = s;
  a_d[idx] = d;
}

// mbits=-inf, denom=0, acc=0
__global__ __launch_bounds__(256) void gat_init(int* __restrict__ mbits,
                                                float* __restrict__ denom,
                                                float* __restrict__ acc, int N) {
  int idx = blockIdx.x * blockDim.x + threadIdx.x;
  if (idx < N * 64) acc[idx] = 0.f;
  if (idx < N * 8) {
    mbits[idx] = 0xFF800000;  // -inf
    denom[idx] = 0.f;
  }
}

__device__ __forceinline__ void atomicMaxF(int* addr, float v) {
  if (v >= 0.f)
    atomicMax(addr, __float_as_int(v));
  else
    atomicMin((unsigned int*)addr, __float_as_uint(v));
}

__device__ __forceinline__ void edge_decode(long long e, const int* ei, int E,
                                            int& s, int& d) {
  if (e < (long long)E) {
    s = ei[e];
    d = ei[(long long)E + e];
  } else {
    s = d = (int)(e - E);  // self loop
  }
}

__device__ __forceinline__ float lrelu(float v) {
  return v > 0.f ? v : v * NEG_SLOPE;
}

// Pass 1: segment max.  Thread per (edge, head).
__global__ __launch_bounds__(256) void gat_edge_max(const int* __restrict__ ei, int E, int N,
                                                    const float* __restrict__ a_s,
                                                    const float* __restrict__ a_d,
                                                    int* __restrict__ mbits) {
  long long t = (long long)blockIdx.x * blockDim.x + threadIdx.x;
  if (t >= (long long)(E + N) * 8) return;
  int h = (int)(t & 7);
  int s, d;
  edge_decode(t >> 3, ei, E, s, d);
  float v = lrelu(a_s[(long long)s * 8 + h] + a_d[(long long)d * 8 + h]);
  atomicMaxF(&mbits[(long long)d * 8 + h], v);
}

// Pass 2: segment sum of exp(e - m[dst]).
__global__ __launch_bounds__(256) void gat_edge_sum(const int* __restrict__ ei, int E, int N,
                                                    const float* __restrict__ a_s,
                                                    const float* __restrict__ a_d,
                                                    const int* __restrict__ mbits,
                                                    float* __restrict__ denom) {
  long long t = (long long)blockIdx.x * blockDim.x + threadIdx.x;
  if (t >= (long long)(E + N) * 8) return;
  int h = (int)(t & 7);
  int s, d;
  edge_decode(t >> 3, ei, E, s, d);
  float v = lrelu(a_s[(long long)s * 8 + h] + a_d[(long long)d * 8 + h]);
  float ex = __expf(v - __int_as_float(mbits[(long long)d * 8 + h]));
  atomicAdd(&denom[(long long)d * 8 + h], ex);
}

// Pass 3: out[dst] += alpha * feat[src].  Thread per (edge, head), 8 channels each.
__global__ __launch_bounds__(256) void gat_edge_agg(const int* __restrict__ ei, int E, int N,
                                                    const float* __restrict__ a_s,
                                                    const float* __restrict__ a_d,
                                                    const int* __restrict__ mbits,
                                                    const float* __restrict__ denom,
                                                    const float* __restrict__ feat,
                                                    float* __restrict__ acc) {
  long long t = (long long)blockIdx.x * blockDim.x + threadIdx.x;
  if (t >= (long long)(E + N) * 8) return;
  int h = (int)(t & 7);
  int s, d;
  edge_decode(t >> 3, ei, E, s, d);
  float v = lrelu(a_s[(long long)s * 8 + h] + a_d[(long long)d * 8 + h]);
  float ex = __expf(v - __int_as_float(mbits[(long long)d * 8 + h]));
  float alpha = ex / (denom[(long long)d * 8 + h] + EPSI);
  const float* fsrc = feat + (long long)s * 64 + h * 8;
  float* adst = acc + (long long)d * 64 + h * 8;
#pragma unroll
  for (int c = 0; c < 8; ++c) atomicAdd(&adst[c], alpha * fsrc[c]);
}

// Layer-1 finalize (concat=True): feat_out = acc + bias1.
__global__ __launch_bounds__(256) void gat_finalize1(const float* __restrict__ acc,
                                                     const float* __restrict__ bias,
                                                     float* __restrict__ out, int N) {
  int idx = blockIdx.x * blockDim.x + threadIdx.x;
  if (idx >= N * 64) return;
  out[idx] = acc[idx] + bias[idx & 63];
}

// Layer-2 finalize (concat=False): out[n,c] = mean_h acc[n,h,c] + bias2[c].
__global__ __launch_bounds__(256) void gat_finalize2(const float* __restrict__ acc,
                                                     const float* __restrict__ bias,
                                                     float* __restrict__ out, int N) {
  int idx = blockIdx.x * blockDim.x + threadIdx.x;  // n*8 + c
  if (idx >= N * 8) return;
  int c = idx & 7;
  const float* a = acc + (long long)(idx >> 3) * 64 + c;
  float sum = 0.f;
#pragma unroll
  for (int h = 0; h < 8; ++h) sum += a[h * 8];
  out[idx] = 0.125f * sum + bias[c];
}

extern "C" void kernel_launch(void* const* d_in, const int* in_sizes, int n_in,
                              void* d_out, int out_size, void* d_ws, size_t ws_size,
                              hipStream_t stream) {
  const float* x        = (const float*)d_in[0];
  const int*   ei       = (const int*)d_in[1];
  const float* W1       = (const float*)d_in[2];
  const float* att_src1 = (const float*)d_in[3];
  const float* att_dst1 = (const float*)d_in[4];
  const float* bias1    = (const float*)d_in[5];
  const float* W2       = (const float*)d_in[6];
  const float* att_src2 = (const float*)d_in[7];
  const float* att_dst2 = (const float*)d_in[8];
  const float* bias2    = (const float*)d_in[9];
  float* out = (float*)d_out;

  const int N = in_sizes[0] / 128;  // 100000
  const int E = in_sizes[1] / 2;    // 3200000

  // Workspace layout (~64 MB, all L2-resident on MI455X's 192MB L2):
  float* P     = (float*)d_ws;            // N*64  features / layer-2 acc
  float* Q     = P + (size_t)N * 64;      // N*64  layer-1 acc / layer-2 features
  float* a_s   = Q + (size_t)N * 64;      // N*8
  float* a_d   = a_s + (size_t)N * 8;     // N*8
  float* denom = a_d + (size_t)N * 8;     // N*8
  int*   mbits = (int*)(denom + (size_t)N * 8);  // N*8

  const int TB = 256;
  const int gN64  = (N * 64 + TB - 1) / TB;
  const int gN8   = (N * 8 + TB - 1) / TB;
  const long long eThreads = (long long)(E + N) * 8;
  const int gE    = (int)((eThreads + TB - 1) / TB);
  const int gGemm = (N + 31) / 32;

  // ---- Layer 1 ----
  gat_gemm_wmma<<<gGemm, TB, 0, stream>>>(x, W1, P, N, 128);
  gat_init<<<gN64, TB, 0, stream>>>(mbits, denom, Q, N);
  gat_attn_coeff<<<gN8, TB, 0, stream>>>(P, att_src1, att_dst1, a_s, a_d, N);
  gat_edge_max<<<gE, TB, 0, stream>>>(ei, E, N, a_s, a_d, mbits);
  gat_edge_sum<<<gE, TB, 0, stream>>>(ei, E, N, a_s, a_d, mbits, denom);
  gat_edge_agg<<<gE, TB, 0, stream>>>(ei, E, N, a_s, a_d, mbits, denom, P, Q);
  gat_finalize1<<<gN64, TB, 0, stream>>>(Q, bias1, P, N);

  // ---- Layer 2 ----
  gat_gemm_wmma<<<gGemm, TB, 0, stream>>>(P, W2, Q, N, 64);
  gat_init<<<gN64, TB, 0, stream>>>(mbits, denom, P, N);
  gat_attn_coeff<<<gN8, TB, 0, stream>>>(Q, att_src2, att_dst2, a_s, a_d, N);
  gat_edge_max<<<gE, TB, 0, stream>>>(ei, E, N, a_s, a_d, mbits);
  gat_edge_sum<<<gE, TB, 0, stream>>>(ei, E, N, a_s, a_d, mbits, denom);
  gat_edge_agg<<<gE, TB, 0, stream>>>(ei, E, N, a_s, a_d, mbits, denom, Q, P);
  gat_finalize2<<<gN8, TB, 0, stream>>>(P, bias2, out, N);
}